// SideWindowBoxFilter_12446815224540
// MI455X (gfx1250) — compile-verified
//
#include <hip/hip_runtime.h>
#include <hip/hip_bf16.h>
#include <stdint.h>

// ---------------- problem geometry ----------------
#define RADIUS  7
#define ITERS   10
#define IMG     2048
#define DOM     (IMG + 2*RADIUS)      // 2062 : once-edge-padded domain the iterations run on
#define GW      2080                  // guard-plane row stride in floats (2080*4B = 65*128B aligned)
#define GH      2076                  // guard-plane rows  (= DOM + 2*RADIUS zero guard ring)
#define TILE    64                    // core tile
#define REG     (TILE + 2*RADIUS)     // 78  : tile + halo region
#define NT      33                    // ceil(DOM / TILE)
#define SSTR    (REG + 3)             // 81  : prefix-plane stride (odd -> conflict-free banking)
#define PLANE   ((size_t)GW * (size_t)GH)   // floats per channel plane

#if defined(__has_builtin)
#  if __has_builtin(__builtin_amdgcn_tensor_load_to_lds)
#    define HAVE_TDM 1
#  else
#    define HAVE_TDM 0
#  endif
#else
#  define HAVE_TDM 0
#endif

typedef uint32_t tdm_v4u __attribute__((ext_vector_type(4)));
typedef int      tdm_v8i __attribute__((ext_vector_type(8)));
typedef int      tdm_v4i __attribute__((ext_vector_type(4)));

// ---------------------------------------------------------------------------
// Init: build padded+guarded state A (edge-padded image surrounded by zeros),
// zero B fully (its guard ring must be 0; interior is overwritten by pass 1).
// 3-D grid: x = column chunk, y = guard row, z = channel -> no divisions,
// coalesced row-major stores.
// ---------------------------------------------------------------------------
__global__ __launch_bounds__(256) void swf_init(const float* __restrict__ img,
                                                float* __restrict__ A,
                                                float* __restrict__ B)
{
    const int gx = (int)(blockIdx.x * 256 + threadIdx.x);
    if (gx >= GW) return;
    const int gy = (int)blockIdx.y;
    const int c  = (int)blockIdx.z;

    float v = 0.0f;
    const int y = gy - RADIUS;            // domain coords (0..DOM)
    const int x = gx - RADIUS;
    if ((unsigned)y < (unsigned)DOM && (unsigned)x < (unsigned)DOM) {
        // U = pad(img, RADIUS, edge):  U[y][x] = img[clamp(y-7)][clamp(x-7)]
        int sy = y - RADIUS; sy = sy < 0 ? 0 : (sy > IMG - 1 ? IMG - 1 : sy);
        int sx = x - RADIUS; sx = sx < 0 ? 0 : (sx > IMG - 1 ? IMG - 1 : sx);
        v = img[((size_t)sy * IMG + sx) * 3 + c];
    }
    const size_t i = (size_t)c * PLANE + (size_t)gy * GW + (size_t)gx;
    A[i] = v;
    B[i] = 0.0f;
}

// ---------------------------------------------------------------------------
// One side-window iteration over the whole padded domain (one channel/tile
// per block).  TDM loads the 78x78 region, 2D prefix-sum in LDS, then each
// core pixel derives all 8 side-window box averages from 16 lattice reads.
// ---------------------------------------------------------------------------
__global__ __launch_bounds__(256) void swf_iter(const float* __restrict__ src,
                                                float* __restrict__ dst,
                                                float* __restrict__ out,
                                                int final_pass)
{
    __shared__ float xs[REG * REG];          // region, packed row-major (TDM layout)
    __shared__ float S[(REG + 1) * SSTR];    // 2D prefix, zero top row / left col

    const int tid = (int)threadIdx.x;
    const int c   = (int)blockIdx.z;
    int oy = (int)blockIdx.y * TILE; if (oy > DOM - TILE) oy = DOM - TILE;  // clamp last tile
    int ox = (int)blockIdx.x * TILE; if (ox > DOM - TILE) ox = DOM - TILE;

    const float* gtile = src + (size_t)c * PLANE + (size_t)oy * GW + (size_t)ox;

#if HAVE_TDM
    if (tid < 32) {   // one wave issues the tensor DMA (uniform scalar branch)
        uint64_t ga = (uint64_t)(uintptr_t)gtile;
        uint32_t lb = (uint32_t)(uintptr_t)&xs[0];     // low 32 bits = LDS byte offset
        tdm_v4u g0;
        g0[0] = 1u;                                    // count=1, user descriptor
        g0[1] = lb;                                    // lds_addr
        g0[2] = (uint32_t)ga;                          // global_addr[31:0]
        g0[3] = (uint32_t)((ga >> 32) & 0x01FFFFFFu) | 0x80000000u; // addr[56:32] | type=2
        tdm_v8i g1;
        g1[0] = (int)(2u << 16);                       // data_size=2 (4B), no flags, mask=0
        g1[1] = (int)(((uint32_t)GW & 0xFFFFu) << 16); // tensor_dim0[15:0]=2080
        g1[2] = (int)((((uint32_t)GW >> 16) & 0xFFFFu) | (((uint32_t)GH & 0xFFFFu) << 16)); // dim0 hi | dim1 lo
        g1[3] = (int)((((uint32_t)GH >> 16) & 0xFFFFu) | ((uint32_t)REG << 16));            // dim1 hi | tile_dim0=78
        g1[4] = (int)REG;                              // tile_dim1=78, tile_dim2=0
        g1[5] = (int)GW;                               // tensor_dim0_stride[31:0]=2080
        g1[6] = 0;                                     // stride hi / dim1_stride lo
        g1[7] = 0;
        tdm_v4i gz = {0, 0, 0, 0};
#  if defined(__clang_major__) && (__clang_major__ >= 23)
        tdm_v8i gz8 = {0, 0, 0, 0, 0, 0, 0, 0};
        __builtin_amdgcn_tensor_load_to_lds(g0, g1, gz, gz, gz8, 0);
#  else
        __builtin_amdgcn_tensor_load_to_lds(g0, g1, gz, gz, 0);
#  endif
#  if __has_builtin(__builtin_amdgcn_s_wait_tensorcnt)
        __builtin_amdgcn_s_wait_tensorcnt(0);
#  else
        asm volatile("s_wait_tensorcnt 0x0" ::: "memory");
#  endif
    }
#else
    for (int i = tid; i < REG * REG; i += 256)
        xs[i] = gtile[(size_t)(i / REG) * GW + (i % REG)];
#endif
    __syncthreads();

    // zero border of prefix plane
    if (tid <= REG) { S[tid] = 0.0f; S[(size_t)tid * SSTR] = 0.0f; }
    // row prefix: S[r+1][j+1] = sum_{j'<=j} xs[r][j']
    if (tid < REG) {
        const float* xr = xs + tid * REG;
        float*       sr = S + (tid + 1) * SSTR + 1;
        float acc = 0.0f;
#pragma unroll 13
        for (int j = 0; j < REG; ++j) { acc += xr[j]; sr[j] = acc; }
    }
    __syncthreads();
    // column prefix in place: S becomes the 2D exclusive-corner prefix
    if (tid < REG) {
        const int j = tid + 1;
        float acc = 0.0f;
#pragma unroll 13
        for (int r = 1; r <= REG; ++r) {
            acc += S[r * SSTR + j];
            S[r * SSTR + j] = acc;
        }
    }
    __syncthreads();

    const float inv64  = 1.0f / 64.0f;    // (1/8)*(1/8)
    const float inv120 = 1.0f / 120.0f;   // (1/8)*(1/15)

    for (int k = 0; k < 16; ++k) {
        int p  = tid + (k << 8);          // 0..4095 core pixels
        int rr = (p >> 6) + RADIUS;       // region row 7..70
        int cc = (p & 63) + RADIUS;       // region col 7..70
        // box rows [a,b) x cols [al,be) = S[b][be]-S[a][be]-S[b][al]+S[a][al]
        // boundaries needed: rows/cols {i-7, i, i+1, i+8}
        const float* SA = S + (rr - 7) * SSTR;
        const float* SB = S + rr * SSTR;
        const float* SC = S + (rr + 1) * SSTR;
        const float* SD = S + (rr + 8) * SSTR;
        const int cA = cc - 7, cB = cc, cC = cc + 1, cD = cc + 8;

        float a0 = SA[cA], a1 = SA[cB], a2 = SA[cC], a3 = SA[cD];
        float b0 = SB[cA], b1 = SB[cB], b2 = SB[cC], b3 = SB[cD];
        float c0 = SC[cA], c1 = SC[cB], c2 = SC[cC], c3 = SC[cD];
        float d0 = SD[cA], d1 = SD[cB], d2 = SD[cC], d3 = SD[cD];

        // column-range differences per row boundary: L=[c-7,c], R=[c,c+7], F=[c-7,c+7]
        float lA = a2 - a0, rA_ = a3 - a1, fA = a3 - a0;
        float lB = b2 - b0, rB_ = b3 - b1, fB = b3 - b0;
        float lC = c2 - c0, rC_ = c3 - c1, fC = c3 - c0;
        float lD = d2 - d0, rD_ = d3 - d1, fD = d3 - d0;

        // vertical L uses rows A..C, R uses B..D, F uses A..D
        float o0 = (lC  - lA ) * inv64;   // (VL,HL)
        float o1 = (rC_ - rA_) * inv64;   // (VL,HR)
        float o2 = (lD  - lB ) * inv64;   // (VR,HL)
        float o3 = (rD_ - rB_) * inv64;   // (VR,HR)
        float o4 = (fC  - fA ) * inv120;  // (VL,HF)
        float o5 = (fD  - fB ) * inv120;  // (VR,HF)
        float o6 = (lD  - lA ) * inv120;  // (VF,HL)
        float o7 = (rD_ - rA_) * inv120;  // (VF,HR)

        float xv = xs[rr * REG + cc];
        float bd = o0 - xv;  float ba = fabsf(bd);
        float dd;
        dd = o1 - xv; if (fabsf(dd) < ba) { ba = fabsf(dd); bd = dd; }
        dd = o2 - xv; if (fabsf(dd) < ba) { ba = fabsf(dd); bd = dd; }
        dd = o3 - xv; if (fabsf(dd) < ba) { ba = fabsf(dd); bd = dd; }
        dd = o4 - xv; if (fabsf(dd) < ba) { ba = fabsf(dd); bd = dd; }
        dd = o5 - xv; if (fabsf(dd) < ba) { ba = fabsf(dd); bd = dd; }
        dd = o6 - xv; if (fabsf(dd) < ba) { ba = fabsf(dd); bd = dd; }
        dd = o7 - xv; if (fabsf(dd) < ba) { ba = fabsf(dd); bd = dd; }
        float nv = xv + bd;

        if (!final_pass) {
            dst[(size_t)c * PLANE + (size_t)(oy + rr) * GW + (size_t)(ox + cc)] = nv;
        } else {
            int y = oy + rr - RADIUS;     // domain coords
            int x = ox + cc - RADIUS;
            if (y >= RADIUS && y < DOM - RADIUS && x >= RADIUS && x < DOM - RADIUS)
                out[((size_t)(y - RADIUS) * IMG + (size_t)(x - RADIUS)) * 3 + c] = nv;
        }
    }
}

// ---------------------------------------------------------------------------
extern "C" void kernel_launch(void* const* d_in, const int* in_sizes, int n_in,
                              void* d_out, int out_size, void* d_ws, size_t ws_size,
                              hipStream_t stream)
{
    const float* img = (const float*)d_in[0];
    // d_in[1]/d_in[2] (kernel1/kernel2) are unused: all taps are uniform box
    // weights (1/8, 1/15), so the convolutions are exact prefix-sum box sums.
    float* out = (float*)d_out;
    float* A = (float*)d_ws;
    float* B = A + 3 * PLANE;

    if (ws_size < 2 * 3 * PLANE * sizeof(float)) return;   // scratch too small

    {
        dim3 g((GW + 255) / 256, GH, 3);
        swf_init<<<g, 256, 0, stream>>>(img, A, B);
    }

    dim3 grid(NT, NT, 3);
    for (int t = 0; t < ITERS; ++t) {
        const float* src = (t & 1) ? B : A;
        float*       dst = (t & 1) ? A : B;
        int fin = (t == ITERS - 1);
        swf_iter<<<grid, 256, 0, stream>>>(src, dst, out, fin);
    }
}